// PermutedSparseWeight_13589276525259
// MI455X (gfx1250) — compile-verified
//
#include <hip/hip_runtime.h>
#include <hip/hip_bf16.h>
#include <math.h>

// CDNA5 / gfx1250: wave32, WMMA f32 16x16x4 for exact-fp32 block-diagonal mixing.

typedef __attribute__((ext_vector_type(2))) float v2f;
typedef __attribute__((ext_vector_type(8))) float v8f;

#define D 4096
#define BLK 64
#define NP 8
#define LS 66  // LDS row stride (pad 64 -> 66: conflict-free A-fragment column reads)

// ---------------------------------------------------------------------------
// Kernel 1: build block-diagonal mixing matrices.
//   A_r[j][j'] = sum_p c0[p, r*64+j] * [perm0[p, r*64+j] == r*64+j']
//   B_c[k'][k] = sum_p c1[p, c*64+k] * [perm1[p, c*64+k] == c*64+k']
// blocks 0..63 build A blocks, 64..127 build B blocks. Row/col ownership per
// thread => no write races.
// ---------------------------------------------------------------------------
__global__ void __launch_bounds__(256) build_mix_kernel(
    const float* __restrict__ c0, const float* __restrict__ c1,
    const int* __restrict__ p0, const int* __restrict__ p1,
    float* __restrict__ wsA, float* __restrict__ wsB) {
  __shared__ float tile[BLK * BLK];
  const int blk = blockIdx.x;  // 0..127
  const int t = threadIdx.x;   // 0..255

#pragma unroll
  for (int g = 0; g < 16; ++g) tile[t + 256 * g] = 0.0f;
  __syncthreads();

  if (blk < 64) {
    if (t < BLK) {
      const int base = blk * BLK + t;  // global row j
#pragma unroll
      for (int p = 0; p < NP; ++p) {
        const int jp = p0[p * D + base] & (BLK - 1);  // block-local target
        tile[t * BLK + jp] += c0[p * D + base];
      }
    }
  } else {
    if (t < BLK) {
      const int base = (blk - 64) * BLK + t;  // global col k
#pragma unroll
      for (int p = 0; p < NP; ++p) {
        const int kp = p1[p * D + base] & (BLK - 1);
        tile[kp * BLK + t] += c1[p * D + base];
      }
    }
  }
  __syncthreads();

  float* dst = (blk < 64) ? (wsA + (size_t)blk * BLK * BLK)
                          : (wsB + (size_t)(blk - 64) * BLK * BLK);
#pragma unroll
  for (int g = 0; g < 16; ++g) dst[t + 256 * g] = tile[t + 256 * g];
}

// ---------------------------------------------------------------------------
// 64x64x64 fp32 matmul helper on LDS tiles using V_WMMA_F32_16X16X4_F32.
// A-operand layout (ISA 7.12.2): lanes 0-15 hold M=0..15 with K={k0,k0+1},
// lanes 16-31 hold M=0..15 with K={k0+2,k0+3}. B mirrors with N across lanes.
// ---------------------------------------------------------------------------
__device__ __forceinline__ v8f mm64(const float* __restrict__ L,
                                    const float* __restrict__ R,
                                    int mi, int ni, int lane) {
  v8f acc = {};
  const int half = lane >> 4;
  const int m = lane & 15;
  const int arow = mi * 16 + m;
  const int bcol = ni * 16 + m;
#pragma unroll
  for (int k0 = 0; k0 < BLK; k0 += 4) {
    const int ka = k0 + half * 2;
    v2f a, b;
    a.x = L[arow * LS + ka];
    a.y = L[arow * LS + ka + 1];
    b.x = R[ka * LS + bcol];
    b.y = R[(ka + 1) * LS + bcol];
    acc = __builtin_amdgcn_wmma_f32_16x16x4_f32(
        /*neg_a=*/false, a, /*neg_b=*/false, b,
        /*c_mod=*/(short)0, acc, /*reuse_a=*/false, /*reuse_b=*/false);
  }
  return acc;
}

// ---------------------------------------------------------------------------
// Kernel 2: one workgroup per 64x64 tile.
//   W  = topk-2:4(|X|) masked X tile           (VALU, into LDS)
//   T  = W @ B_c                               (WMMA fp32, T overwrites sW)
//   Out= A_r @ T                               (WMMA fp32, direct store)
// LDS: 3 tiles * 64*66*4B = 50.7 KB.
// ---------------------------------------------------------------------------
__global__ void __launch_bounds__(256) permuted_sparse_wmma_kernel(
    const float* __restrict__ X, const float* __restrict__ wsA,
    const float* __restrict__ wsB, float* __restrict__ out) {
  __shared__ float sW[BLK * LS];  // masked X, later reused to hold T
  __shared__ float sB[BLK * LS];
  __shared__ float sA[BLK * LS];

  const int tileId = blockIdx.x;
  const int rb = tileId >> 6;   // row block 0..63
  const int cb = tileId & 63;   // col block 0..63
  const int t = threadIdx.x;

  // ---- load X tile, apply 2:4 mask (keep 2 largest |x| per group of 4;
  //      stable-argsort tie-break: equal magnitudes prefer higher index) ----
#pragma unroll
  for (int g = 0; g < 4; ++g) {
    const int gid = t + g * 256;           // 0..1023 groups of 4
    const int row = gid >> 4;
    const int j0 = (gid & 15) * 4;
    const float4 x4 = *(const float4*)(X + (size_t)(rb * BLK + row) * D +
                                       cb * BLK + j0);
    const float vv[4] = {x4.x, x4.y, x4.z, x4.w};
    const float av[4] = {fabsf(x4.x), fabsf(x4.y), fabsf(x4.z), fabsf(x4.w)};
#pragma unroll
    for (int e = 0; e < 4; ++e) {
      int rank = 0;
#pragma unroll
      for (int s = 0; s < 4; ++s)
        rank += (av[s] > av[e]) || ((av[s] == av[e]) && (s > e));
      sW[row * LS + j0 + e] = (rank < 2) ? vv[e] : 0.0f;
    }
  }

  // ---- load mixing tiles A_r, B_c ----
#pragma unroll
  for (int g = 0; g < 16; ++g) {
    const int idx = t + g * 256;  // 0..4095
    const int r = idx >> 6, c = idx & 63;
    sB[r * LS + c] = wsB[(size_t)cb * BLK * BLK + idx];
    sA[r * LS + c] = wsA[(size_t)rb * BLK * BLK + idx];
  }
  __syncthreads();

  const int lane = t & 31;
  const int wid = t >> 5;       // 8 waves, 2 subtiles each
  const int half = lane >> 4;
  const int m = lane & 15;

  // ---- phase 1: T = W @ B ----
  v8f t0, t1;
  {
    const int s0 = wid * 2, s1 = wid * 2 + 1;
    t0 = mm64(sW, sB, s0 >> 2, s0 & 3, lane);
    t1 = mm64(sW, sB, s1 >> 2, s1 & 3, lane);
  }
  __syncthreads();  // all reads of sW done -> safe to overwrite with T
#pragma unroll
  for (int j = 0; j < 2; ++j) {
    const int s = wid * 2 + j;
    const int mi = s >> 2, ni = s & 3;
    const v8f acc = j ? t1 : t0;
#pragma unroll
    for (int v = 0; v < 8; ++v)
      sW[(mi * 16 + 8 * half + v) * LS + ni * 16 + m] = acc[v];
  }
  __syncthreads();

  // ---- phase 2: Out = A @ T, store fp32 ----
#pragma unroll
  for (int j = 0; j < 2; ++j) {
    const int s = wid * 2 + j;
    const int mi = s >> 2, ni = s & 3;
    const v8f acc = mm64(sA, sW, mi, ni, lane);
    const size_t base = (size_t)(rb * BLK + mi * 16 + 8 * half) * D +
                        cb * BLK + ni * 16 + m;
#pragma unroll
    for (int v = 0; v < 8; ++v) out[base + (size_t)v * D] = acc[v];
  }
}

// ---------------------------------------------------------------------------
extern "C" void kernel_launch(void* const* d_in, const int* in_sizes, int n_in,
                              void* d_out, int out_size, void* d_ws,
                              size_t ws_size, hipStream_t stream) {
  const float* X  = (const float*)d_in[0];   // [4096,4096] f32
  const float* c0 = (const float*)d_in[1];   // [8,4096] f32
  const float* c1 = (const float*)d_in[2];   // [8,4096] f32
  // d_in[3] = mask (bool) -- recomputed in-kernel from X (deterministic)
  const int* p0 = (const int*)d_in[4];       // [8,4096] i32
  const int* p1 = (const int*)d_in[5];       // [8,4096] i32
  float* out = (float*)d_out;

  float* wsA = (float*)d_ws;                 // 64 blocks * 64*64 f32 = 1 MB
  float* wsB = wsA + (size_t)64 * BLK * BLK; // 1 MB

  build_mix_kernel<<<128, 256, 0, stream>>>(c0, c1, p0, p1, wsA, wsB);
  permuted_sparse_wmma_kernel<<<(D / BLK) * (D / BLK), 256, 0, stream>>>(
      X, wsA, wsB, out);
}